// LinearEmbedded_45157286150934
// MI455X (gfx1250) — compile-verified
//
#include <hip/hip_runtime.h>

typedef __attribute__((ext_vector_type(2))) float v2f;
typedef __attribute__((ext_vector_type(8))) float v8f;
typedef __attribute__((ext_vector_type(4))) int   v4i;

#define A_DIM 128
#define B_DIM 128
#define C_DIM 512
#define D_DIM 512
#define KC    32               // K-chunk staged in LDS
#define DT    128              // D columns per block
#define XS    36               // lds_x row stride (floats): 144B rows, 16B-aligned (async B128), conflict-free b64 reads
#define WPAD  34               // lds_wt row stride (floats): 136B rows, b64-aligned, conflict-free frag reads
#define NCHUNK (C_DIM / KC)    // 16
#define NBUF  3                // triple buffer: iter-c writes hit buf[(c+1)%3]; last readers were iter c-2

#define LDS_AS __attribute__((address_space(3)))
#define GLB_AS __attribute__((address_space(1)))

#if __has_builtin(__builtin_amdgcn_global_load_async_to_lds_b128) && \
    __has_builtin(__builtin_amdgcn_s_wait_asynccnt)
#define USE_ASYNC 1
#else
#define USE_ASYNC 0
#endif

__global__ __launch_bounds__(256)
void LinearEmbedded_45157286150934_kernel(const float* __restrict__ x,
                                          const int*   __restrict__ region_ix,
                                          const float* __restrict__ w,
                                          const float* __restrict__ bias,
                                          float*       __restrict__ out)
{
    __shared__ float lds_x [NBUF][A_DIM * XS];    // [buf][row a][k]
    __shared__ float lds_wt[NBUF][DT    * WPAD];  // [buf][d_local][k] (transposed)

    const int b    = blockIdx.x;                  // region / minibatch index
    const int d0   = blockIdx.y * DT;             // D-tile base
    const int r    = region_ix[b];
    const int tid  = threadIdx.x;
    const int lane = tid & 31;
    const int wv   = tid >> 5;                    // wave 0..7
    const int l15  = lane & 15;
    const int sel  = lane >> 4;                   // 0: K pair {0,1}; 1: K pair {2,3}
    const int rowgrp = wv & 3;                    // 4 row groups x 32 A-rows
    const int colgrp = wv >> 2;                   // 2 col groups x 64 D-cols

    const float* xg = x + (size_t)b * C_DIM;                          // + a*B*C + c
    const float* wg = w + (size_t)r * ((size_t)C_DIM * D_DIM) + d0;   // + k*D + d_local

    v8f acc[2][4];
    #pragma unroll
    for (int mt = 0; mt < 2; ++mt)
        #pragma unroll
        for (int jt = 0; jt < 4; ++jt)
            acc[mt][jt] = (v8f){0.f,0.f,0.f,0.f,0.f,0.f,0.f,0.f};

    float4 pw[4];   // register staging for the W chunk (transposed on LDS store)

    // ---- x chunk: async DMA straight into LDS buffer p (4 x 16B per thread) ----
    auto issue_x = [&](int kc, int p) {
        #pragma unroll
        for (int i = 0; i < 4; ++i) {
            int idx = tid + 256 * i;              // 0..1023
            int row = idx >> 3, c4 = idx & 7;     // 128 rows x 8 float4
#if USE_ASYNC
            __builtin_amdgcn_global_load_async_to_lds_b128(
                (GLB_AS v4i*)(xg + (size_t)row * (B_DIM * C_DIM) + kc + c4 * 4),
                (LDS_AS v4i*)&lds_x[p][row * XS + c4 * 4], 0, 0);
#else
            float4 vx = *(const float4*)(xg + (size_t)row * (B_DIM * C_DIM) + kc + c4 * 4);
            *(float4*)&lds_x[p][row * XS + c4 * 4] = vx;
#endif
        }
    };
    // ---- W chunk: issue global loads into registers (drained after compute) ----
    auto g_load_w = [&](int kc) {
        #pragma unroll
        for (int i = 0; i < 4; ++i) {
            int idx = tid + 256 * i;
            int k = idx >> 5, d4 = idx & 31;      // 32 k-rows x 32 float4 along d
            pw[i] = *(const float4*)(wg + (size_t)(kc + k) * D_DIM + d4 * 4);
        }
    };
    auto l_store_w = [&](int p) {                 // transposed drain: lds_wt[d][k]
        #pragma unroll
        for (int i = 0; i < 4; ++i) {
            int idx = tid + 256 * i;
            int k = idx >> 5, d4 = idx & 31;
            lds_wt[p][(d4 * 4 + 0) * WPAD + k] = pw[i].x;
            lds_wt[p][(d4 * 4 + 1) * WPAD + k] = pw[i].y;
            lds_wt[p][(d4 * 4 + 2) * WPAD + k] = pw[i].z;
            lds_wt[p][(d4 * 4 + 3) * WPAD + k] = pw[i].w;
        }
    };

    // ---- 8 K-steps x 8 WMMAs per chunk on buffer p (all-b64 fragment reads) ----
    auto compute = [&](int p) {
        #pragma unroll
        for (int kk = 0; kk < KC; kk += 4) {
            const int ko = kk + 2 * sel;
            v2f af[2], bf[4];
            #pragma unroll
            for (int mt = 0; mt < 2; ++mt)
                af[mt] = *(const v2f*)&lds_x[p][(rowgrp * 32 + mt * 16 + l15) * XS + ko];
            #pragma unroll
            for (int jt = 0; jt < 4; ++jt)
                bf[jt] = *(const v2f*)&lds_wt[p][(colgrp * 64 + jt * 16 + l15) * WPAD + ko];
            #pragma unroll
            for (int mt = 0; mt < 2; ++mt)
                #pragma unroll
                for (int jt = 0; jt < 4; ++jt)
                    acc[mt][jt] = __builtin_amdgcn_wmma_f32_16x16x4_f32(
                        false, af[mt], false, bf[jt],
                        (short)0, acc[mt][jt], false, false);
        }
    };

    // ---- pipeline ----
    issue_x(0, 0);
    g_load_w(0);
    l_store_w(0);
    for (int c = 0; c < NCHUNK; ++c) {
        if (c + 1 < NCHUNK) {
            issue_x((c + 1) * KC, (c + 1) % NBUF);   // x DMA overlaps WMMAs below
            g_load_w((c + 1) * KC);                  // W HBM fetch overlaps WMMAs below
#if USE_ASYNC
            __builtin_amdgcn_s_wait_asynccnt(4);     // x chunk c landed; chunk c+1 in flight
#endif
        } else {
#if USE_ASYNC
            __builtin_amdgcn_s_wait_asynccnt(0);
#endif
        }
        __syncthreads();                             // chunk c fully visible (x async + W ds_stores)
        compute(c % NBUF);
        if (c + 1 < NCHUNK)
            l_store_w((c + 1) % NBUF);               // drain W regs after compute (loadcnt hidden)
    }

    // ---- epilogue: add gathered bias, store fp32 ----
    #pragma unroll
    for (int jt = 0; jt < 4; ++jt) {
        int dcol = d0 + colgrp * 64 + jt * 16 + l15;
        float bv = bias[(size_t)r * D_DIM + dcol];
        #pragma unroll
        for (int mt = 0; mt < 2; ++mt) {
            #pragma unroll
            for (int vi = 0; vi < 8; ++vi) {
                int a_row = rowgrp * 32 + mt * 16 + vi + 8 * sel;  // C/D layout: VGPR vi -> M = vi + 8*sel
                out[(size_t)a_row * (B_DIM * D_DIM) + (size_t)b * D_DIM + dcol]
                    = acc[mt][jt][vi] + bv;
            }
        }
    }
}

extern "C" void kernel_launch(void* const* d_in, const int* in_sizes, int n_in,
                              void* d_out, int out_size, void* d_ws, size_t ws_size,
                              hipStream_t stream)
{
    const float* x         = (const float*)d_in[0];
    const int*   region_ix = (const int*)  d_in[1];
    const float* weight1   = (const float*)d_in[2];
    const float* bias1     = (const float*)d_in[3];
    float*       out       = (float*)d_out;

    dim3 grid(B_DIM, D_DIM / DT);   // 128 regions x 4 D-tiles
    dim3 block(256);                // 8 wave32
    LinearEmbedded_45157286150934_kernel<<<grid, block, 0, stream>>>(
        x, region_ix, weight1, bias1, out);
}